// CrossEntropyLoss_18107582120438
// MI455X (gfx1250) — compile-verified
//
#include <hip/hip_runtime.h>

// Cross-entropy loss for logits (16, 513, 32000) f32, targets/lengths int64.
// HBM-bound: ~1.05 GB streamed once -> ~45us floor at 23.3 TB/s on MI455X.
// One workgroup per (b, s) row; online softmax in registers; wave32 shuffles;
// deterministic 2-stage global reduction; WMMA f32 16x16x4 (ones-B) used for
// the block-sum epilogue in the finalize kernel.

#define NROWS 8192      // 16 * 512
#define VDIM  32000
#define SP1   513

typedef float fv4 __attribute__((ext_vector_type(4)));
typedef float v2f __attribute__((ext_vector_type(2)));
typedef float v8f __attribute__((ext_vector_type(8)));

#define L2E 1.4426950408889634f   // log2(e)
#define LN2 0.6931471805599453f   // ln(2)

__device__ __forceinline__ float fast_exp2(float x) {
#if __has_builtin(__builtin_amdgcn_exp2f)
  return __builtin_amdgcn_exp2f(x);       // v_exp_f32
#else
  return exp2f(x);
#endif
}

__device__ __forceinline__ float fast_log2(float x) {
#if __has_builtin(__builtin_amdgcn_logf)
  return __builtin_amdgcn_logf(x);        // v_log_f32 (base-2)
#else
  return log2f(x);
#endif
}

// Combine two online-softmax partials (m in log2-units, s = sum of 2^(t - m)).
__device__ __forceinline__ void combine(float& ml, float& s, float mo, float so) {
  float M = fmaxf(ml, mo);
  s = s * fast_exp2(ml - M) + so * fast_exp2(mo - M);
  ml = M;
}

__global__ __launch_bounds__(256)
void ce_row_kernel(const float* __restrict__ logits,      // (16, 513, 32000)
                   const long long* __restrict__ trg,     // (16, 513)
                   const long long* __restrict__ lengths, // (16,)
                   float* __restrict__ ws_nll,            // (8192,)
                   float* __restrict__ ws_cnt)            // (8192,)
{
  const int row = blockIdx.x;          // 0..8191
  const int b   = row >> 9;            // row / 512
  const int j   = row & 511;           // position within logits[:, 1:]
  const float* base = logits + ((size_t)b * SP1 + (size_t)(j + 1)) * (size_t)VDIM;
  const fv4* p4 = (const fv4*)base;    // 128-bit aligned (row stride = 128000 B)
  const int tid = threadIdx.x;

  // --- single streaming pass: per-lane online (max, sum-exp2) ---
  float ml = -__builtin_inff();        // running max, in log2 units (x * log2e)
  float s  = 0.0f;                     // sum of 2^(x*log2e - ml)
  for (int i = tid; i < VDIM / 4; i += 256) {
    fv4 x = __builtin_nontemporal_load(&p4[i]);   // global_load_b128, TH=NT
    float c = fmaxf(fmaxf(x.x, x.y), fmaxf(x.z, x.w)) * L2E;
    float M = fmaxf(ml, c);
    s = s * fast_exp2(ml - M);         // rescale (exp2(0)=1 when no update)
    ml = M;
    s += fast_exp2(fmaf(x.x, L2E, -M)) + fast_exp2(fmaf(x.y, L2E, -M))
       + fast_exp2(fmaf(x.z, L2E, -M)) + fast_exp2(fmaf(x.w, L2E, -M));
  }

  // --- wave32 reduction of (ml, s) ---
#pragma unroll
  for (int k = 16; k >= 1; k >>= 1) {
    float mo = __shfl_xor(ml, k, 32);
    float so = __shfl_xor(s,  k, 32);
    combine(ml, s, mo, so);
  }

  // --- cross-wave (8 waves) via LDS ---
  __shared__ float rm[8];
  __shared__ float rs[8];
  const int lane = tid & 31, wid = tid >> 5;
  if (lane == 0) { rm[wid] = ml; rs[wid] = s; }
  __syncthreads();

  if (tid < 8) {
    ml = rm[tid];
    s  = rs[tid];
#pragma unroll
    for (int k = 4; k >= 1; k >>= 1) {       // lanes 0..7 only (width 8)
      float mo = __shfl_xor(ml, k, 8);
      float so = __shfl_xor(s,  k, 8);
      combine(ml, s, mo, so);
    }
    if (tid == 0) {
      const long long tgt = trg[(size_t)b * SP1 + (size_t)(j + 1)];
      const float xt  = base[tgt];                       // gather target logit
      const float lse = LN2 * (ml + fast_log2(s));       // logsumexp of row
      const float nll = lse - xt;
      const bool valid = ((long long)j < lengths[b]) && (tgt != 0);
      ws_nll[row] = valid ? nll : 0.0f;
      ws_cnt[row] = valid ? 1.0f : 0.0f;
    }
  }
}

// Reduce one lane-value across the whole wave on the matrix pipe:
// A(16x4) x ones(4x16): A.vgpr0 = v (K0 lanes0-15 / K2 lanes16-31), A.vgpr1 = 0.
// D[m][n] = v[m] + v[m+16]; per-lane sum of the 8 D VGPRs + shfl_xor(16) gives
// the full 32-lane sum in every lane. Requires EXEC all-ones (it is here).
__device__ __forceinline__ float wave_sum(float v) {
#if __has_builtin(__builtin_amdgcn_wmma_f32_16x16x4_f32)
  v2f A = {v, 0.0f};
  v2f B = {1.0f, 1.0f};
  v8f C = {0.f, 0.f, 0.f, 0.f, 0.f, 0.f, 0.f, 0.f};
  C = __builtin_amdgcn_wmma_f32_16x16x4_f32(
      /*neg_a=*/false, A, /*neg_b=*/false, B,
      /*c_mod=*/(short)0, C, /*reuse_a=*/false, /*reuse_b=*/false);
  float t = C[0] + C[1] + C[2] + C[3] + C[4] + C[5] + C[6] + C[7];
  return t + __shfl_xor(t, 16, 32);
#else
  float t = v;
#pragma unroll
  for (int k = 16; k >= 1; k >>= 1) t += __shfl_xor(t, k, 32);
  return t;
#endif
}

__global__ __launch_bounds__(256)
void ce_finalize_kernel(const float* __restrict__ ws_nll,
                        const float* __restrict__ ws_cnt,
                        float* __restrict__ out)
{
  const int tid = threadIdx.x;
  float a = 0.0f, c = 0.0f;
#pragma unroll
  for (int i = 0; i < NROWS / 256; ++i) {      // fixed order -> deterministic
    const int idx = tid + i * 256;
    a += ws_nll[idx];
    c += ws_cnt[idx];
  }

  const float wa = wave_sum(a);                // v_wmma_f32_16x16x4_f32
  const float wc = wave_sum(c);

  __shared__ float sa[8];
  __shared__ float sc[8];
  const int lane = tid & 31, wid = tid >> 5;
  if (lane == 0) { sa[wid] = wa; sc[wid] = wc; }
  __syncthreads();

  if (tid == 0) {
    float A = 0.0f, C = 0.0f;
#pragma unroll
    for (int w = 0; w < 8; ++w) { A += sa[w]; C += sc[w]; }
    out[0] = A / fmaxf(C, 1.0f);
  }
}

extern "C" void kernel_launch(void* const* d_in, const int* in_sizes, int n_in,
                              void* d_out, int out_size, void* d_ws, size_t ws_size,
                              hipStream_t stream) {
  (void)in_sizes; (void)n_in; (void)out_size; (void)ws_size;
  const float*     logits  = (const float*)d_in[0];     // (16, 513, 32000) f32
  const long long* trg     = (const long long*)d_in[1]; // (16, 513) i64
  const long long* lengths = (const long long*)d_in[2]; // (16,) i64

  float* ws_nll = (float*)d_ws;          // 8192 floats
  float* ws_cnt = ws_nll + NROWS;        // 8192 floats  (64 KB total)

  ce_row_kernel<<<NROWS, 256, 0, stream>>>(logits, trg, lengths, ws_nll, ws_cnt);
  ce_finalize_kernel<<<1, 256, 0, stream>>>(ws_nll, ws_cnt, (float*)d_out);
}